// KOP2d_25658134626403
// MI455X (gfx1250) — compile-verified
//
#include <hip/hip_runtime.h>

// ---- problem constants (match reference) ----
constexpr int N_   = 128;
constexpr int LOGN = 7;
constexpr int NP   = N_ * N_;      // 16384 pixels
constexpr int Bb   = 64;
constexpr int CIN  = 16;
constexpr int COUT = 16;

typedef float v2f __attribute__((ext_vector_type(2)));
typedef float v8f __attribute__((ext_vector_type(8)));

// ---- CDNA5 async global->LDS copy (ASYNCcnt path), guarded for portability ----
#if defined(__has_builtin)
#if __has_builtin(__builtin_amdgcn_global_load_async_to_lds_b128)
#define HAVE_ASYNC_LDS 1
#endif
#endif

#ifdef HAVE_ASYNC_LDS
// builtin expects: (v4i addrspace(1)* src, v4i addrspace(3)* dst, imm offset, imm cpol)
typedef int v4i __attribute__((__vector_size__(4 * sizeof(int))));
typedef __attribute__((address_space(1))) v4i g_v4i;
typedef __attribute__((address_space(3))) v4i l_v4i;

static __device__ __forceinline__ void async_wait0() {
#if __has_builtin(__builtin_amdgcn_s_wait_asynccnt)
    __builtin_amdgcn_s_wait_asynccnt(0);
#else
    asm volatile("s_wait_asynccnt 0x0" ::: "memory");
#endif
}
#endif

// ---------------------------------------------------------------------------
// Butterfly tensor-product kernel: one workgroup per 128x128 complex image.
// Whole image lives in 128KB of dynamic LDS (WGP has 320KB).
//   in_mode : 0 = complex image-major (float2)  [async global->LDS B128 path]
//             1 = real image-major (pack imag=0)          [for x]
//             2 = real 3x3 kernel, zero-pad to 128x128     [for weight]
//   out_mode: 0 = complex image-major
//             1 = pixel-major X  [p][b][i]   (img = b*CIN + i)
//             2 = pixel-major W  [p][i][o]   (img = o*CIN + i)
//             3 = real image-major (take .re)              [final output]
// tw layout: [7][64][2][2][2] floats (stage, pair, 2out, 2in, re/im)
// ---------------------------------------------------------------------------
__global__ void butterfly_tp_kernel(const float* __restrict__ in,
                                    const float* __restrict__ tw1,   // along W
                                    const float* __restrict__ tw2,   // along H
                                    float*       __restrict__ out,
                                    int in_mode, int out_mode, int increasing)
{
    extern __shared__ float2 tile[];   // [128][128] complex = 128 KB
    const int img = blockIdx.x;
    const int tid = threadIdx.x;
    const int nth = blockDim.x;

    // Twiddles are reused by every block: nudge them toward L2 early.
    __builtin_prefetch(tw1, 0, 1);
    __builtin_prefetch(tw2, 0, 1);

    // ---- load image into LDS ----
    if (in_mode == 0) {
        const float2* src = (const float2*)in + (size_t)img * NP;
#ifdef HAVE_ASYNC_LDS
        // 16B chunks, DMA'd straight into LDS (no VGPR staging), ASYNCcnt-tracked.
        for (int e4 = tid; e4 < NP / 2; e4 += nth) {
            __builtin_amdgcn_global_load_async_to_lds_b128(
                (g_v4i*)(src + 2 * e4), (l_v4i*)(tile + 2 * e4), 0, 0);
        }
        async_wait0();   // each wave waits for its own async loads
#else
        for (int e = tid; e < NP; e += nth) tile[e] = src[e];
#endif
    } else if (in_mode == 1) {
        const float* src = in + (size_t)img * NP;
        for (int e = tid; e < NP; e += nth) tile[e] = make_float2(src[e], 0.0f);
    } else { // 3x3 kernel zero-padded, img = o*CIN + i
        for (int e = tid; e < NP; e += nth) {
            int h = e >> 7, w = e & 127;
            float v = 0.0f;
            if (h < 3 && w < 3) v = in[((size_t)img * 3 + h) * 3 + w];
            tile[e] = make_float2(v, 0.0f);
        }
    }
    __syncthreads();

    // ---- 7 stages along W (within rows), twiddles tw1 ----
    for (int idx = 0; idx < LOGN; ++idx) {
        const int sh = increasing ? idx : (LOGN - 1 - idx);
        const int stride = 1 << sh;
        const float* t = tw1 + (size_t)idx * 64 * 8;
        for (int q = tid; q < 64 * N_; q += nth) {
            const int k = q & 63;        // pair index (== gi*stride + s)
            const int r = q >> 6;        // row
            const int gi = k >> sh;
            const int s  = k & (stride - 1);
            const int p0 = gi * (2 * stride) + s;
            const int p1 = p0 + stride;
            const float* tk = t + k * 8;
            float2 a = tile[r * N_ + p0];
            float2 b = tile[r * N_ + p1];
            float2 o0, o1;
            o0.x = tk[0]*a.x - tk[1]*a.y + tk[2]*b.x - tk[3]*b.y;
            o0.y = tk[0]*a.y + tk[1]*a.x + tk[2]*b.y + tk[3]*b.x;
            o1.x = tk[4]*a.x - tk[5]*a.y + tk[6]*b.x - tk[7]*b.y;
            o1.y = tk[4]*a.y + tk[5]*a.x + tk[6]*b.y + tk[7]*b.x;
            tile[r * N_ + p0] = o0;      // pairs are disjoint per thread/stage
            tile[r * N_ + p1] = o1;
        }
        __syncthreads();
    }

    // ---- 7 stages along H (within columns), twiddles tw2 ----
    for (int idx = 0; idx < LOGN; ++idx) {
        const int sh = increasing ? idx : (LOGN - 1 - idx);
        const int stride = 1 << sh;
        const float* t = tw2 + (size_t)idx * 64 * 8;
        for (int q = tid; q < 64 * N_; q += nth) {
            const int c = q & 127;       // column (fastest -> conflict-free)
            const int k = q >> 7;        // pair index
            const int gi = k >> sh;
            const int s  = k & (stride - 1);
            const int p0 = gi * (2 * stride) + s;
            const int p1 = p0 + stride;
            const float* tk = t + k * 8;
            float2 a = tile[p0 * N_ + c];
            float2 b = tile[p1 * N_ + c];
            float2 o0, o1;
            o0.x = tk[0]*a.x - tk[1]*a.y + tk[2]*b.x - tk[3]*b.y;
            o0.y = tk[0]*a.y + tk[1]*a.x + tk[2]*b.y + tk[3]*b.x;
            o1.x = tk[4]*a.x - tk[5]*a.y + tk[6]*b.x - tk[7]*b.y;
            o1.y = tk[4]*a.y + tk[5]*a.x + tk[6]*b.y + tk[7]*b.x;
            tile[p0 * N_ + c] = o0;
            tile[p1 * N_ + c] = o1;
        }
        __syncthreads();
    }

    // ---- store ----
    if (out_mode == 0) {
        float2* dst = (float2*)out + (size_t)img * NP;
        for (int e = tid; e < NP; e += nth) dst[e] = tile[e];
    } else if (out_mode == 1) {
        const int b = img / CIN, i = img % CIN;
        float2* dst = (float2*)out;
        for (int e = tid; e < NP; e += nth)
            dst[(size_t)e * (Bb * CIN) + b * CIN + i] = tile[e];
    } else if (out_mode == 2) {
        const int o = img / CIN, i = img % CIN;
        float2* dst = (float2*)out;
        for (int e = tid; e < NP; e += nth)
            dst[(size_t)e * (CIN * COUT) + i * COUT + o] = tile[e];
    } else {
        float* dst = out + (size_t)img * NP;
        for (int e = tid; e < NP; e += nth) dst[e] = tile[e].x;
    }
}

// ---------------------------------------------------------------------------
// Per-pixel complex channel GEMM via f32 WMMA (16x16x4).
//   xP  : [p][b(64)][i(16)] complex
//   wP  : [p][i(16)][o(16)] complex
//   prod: [(b*COUT+o)][p]    complex   (image-major for the k2 transform)
// One block (4 waves) per pixel; wave w computes M-tile w of the 64x16 result.
// Complex GEMM: Re = Xr·Wr + (-Xi)·Wi ; Im = Xr·Wi + Xi·Wr  (two f32 accums).
// ---------------------------------------------------------------------------
__global__ void pixel_cgemm_wmma(const float2* __restrict__ xP,
                                 const float2* __restrict__ wP,
                                 float2*       __restrict__ prod)
{
    const int p    = blockIdx.x;
    const int lane = threadIdx.x & 31;
    const int wave = threadIdx.x >> 5;            // 0..3 -> M tile

    const float2* X = xP + (size_t)p * (Bb * CIN);
    const float2* W = wP + (size_t)p * (CIN * COUT);

    const int mrow  = wave * 16 + (lane & 15);    // A row (batch index)
    const int khalf = (lane >> 4) * 2;            // lanes 16-31 hold K+2/K+3
    const int ncol  = lane & 15;                  // B/C column (cout index)

    v8f accRe = {};
    v8f accIm = {};

    for (int kc = 0; kc < 4; ++kc) {              // K = 16 in chunks of 4
        const int k0 = kc * 4 + khalf;
        // A fragment (16x4 f32): VGPR0 = K=k0, VGPR1 = K=k0+1 for this lane
        const float2 xa0 = X[mrow * CIN + k0];
        const float2 xa1 = X[mrow * CIN + k0 + 1];
        v2f ar  = {  xa0.x,  xa1.x };
        v2f ai  = {  xa0.y,  xa1.y };
        v2f nai = { -xa0.y, -xa1.y };             // f32 WMMA has no A-neg bit
        // B fragment (4x16 f32): row-striped across lanes
        const float2 wb0 = W[(k0)     * COUT + ncol];
        const float2 wb1 = W[(k0 + 1) * COUT + ncol];
        v2f br = { wb0.x, wb1.x };
        v2f bi = { wb0.y, wb1.y };

        accRe = __builtin_amdgcn_wmma_f32_16x16x4_f32(false, ar,  false, br,
                                                      (short)0, accRe, false, false);
        accRe = __builtin_amdgcn_wmma_f32_16x16x4_f32(false, nai, false, bi,
                                                      (short)0, accRe, false, false);
        accIm = __builtin_amdgcn_wmma_f32_16x16x4_f32(false, ar,  false, bi,
                                                      (short)0, accIm, false, false);
        accIm = __builtin_amdgcn_wmma_f32_16x16x4_f32(false, ai,  false, br,
                                                      (short)0, accIm, false, false);
    }

    // C/D layout: VGPR r holds (M = (lane>>4)*8 + r, N = lane&15)
    const int o = ncol;
#pragma unroll
    for (int r = 0; r < 8; ++r) {
        const int b = wave * 16 + (lane >> 4) * 8 + r;
        prod[(size_t)(b * COUT + o) * NP + p] = make_float2(accRe[r], accIm[r]);
    }
}

// ---------------------------------------------------------------------------
// Launch: x-transform -> w-transform -> per-pixel WMMA GEMM -> k2-transform.
// Workspace (floats):
//   xP   : NP*Bb*CIN*2  = 33,554,432   (134.2 MB)
//   wP   : NP*CIN*COUT*2=  8,388,608   ( 33.6 MB)
//   prod : Bb*COUT*NP*2 = 33,554,432   (134.2 MB)
// ---------------------------------------------------------------------------
extern "C" void kernel_launch(void* const* d_in, const int* in_sizes, int n_in,
                              void* d_out, int out_size, void* d_ws, size_t ws_size,
                              hipStream_t stream) {
    const float* x      = (const float*)d_in[0];
    const float* weight = (const float*)d_in[1];
    const float* k1_tw1 = (const float*)d_in[2];
    const float* k1_tw2 = (const float*)d_in[3];
    const float* kd_tw1 = (const float*)d_in[4];
    const float* kd_tw2 = (const float*)d_in[5];
    const float* k2_tw1 = (const float*)d_in[6];
    const float* k2_tw2 = (const float*)d_in[7];
    float* out = (float*)d_out;

    float* ws   = (float*)d_ws;
    float* xP   = ws;
    float* wP   = xP + (size_t)NP * Bb * CIN * 2;
    float* prod = wP + (size_t)NP * CIN * COUT * 2;

    const size_t shmem = (size_t)NP * sizeof(float2);   // 128 KB dynamic LDS

    // K1 (decreasing stride) on x: real -> complex, pixel-major X output
    butterfly_tp_kernel<<<Bb * CIN, 512, shmem, stream>>>(
        x, k1_tw1, k1_tw2, xP, /*in*/1, /*out*/1, /*inc*/0);

    // Kd (decreasing stride) on zero-padded 3x3 kernel, pixel-major W output
    butterfly_tp_kernel<<<COUT * CIN, 512, shmem, stream>>>(
        weight, kd_tw1, kd_tw2, wP, /*in*/2, /*out*/2, /*inc*/0);

    // per-pixel complex channel GEMM (WMMA f32 16x16x4), image-major output
    pixel_cgemm_wmma<<<NP, 128, 0, stream>>>(
        (const float2*)xP, (const float2*)wP, (float2*)prod);

    // K2 (increasing stride) on product, Complex2Real into d_out
    butterfly_tp_kernel<<<Bb * COUT, 512, shmem, stream>>>(
        prod, k2_tw1, k2_tw2, out, /*in*/0, /*out*/3, /*inc*/1);
}